// CSRSparseRetrievalModel_48928267436211
// MI455X (gfx1250) — compile-verified
//
#include <hip/hip_runtime.h>
#include <math.h>

#define VOCAB 30522
#define TOPK  10
#define NB_TOPK 512

typedef __attribute__((ext_vector_type(2))) float v2f;
typedef __attribute__((ext_vector_type(8))) float v8f;

// ---------------------------------------------------------------------------
// 1) zero the dense query vector
// ---------------------------------------------------------------------------
__global__ void zero_query_kernel(float* __restrict__ q) {
    int i = blockIdx.x * blockDim.x + threadIdx.x;
    if (i < VOCAB) q[i] = 0.0f;
}

// ---------------------------------------------------------------------------
// 2) scatter-add the COO query (atomicAdd coalesces duplicate indices,
//    matching torch.sparse_coo_tensor(...).to_dense())
// ---------------------------------------------------------------------------
__global__ void scatter_query_kernel(const int* __restrict__ qi,
                                     const float* __restrict__ qv,
                                     float* __restrict__ q, int qnnz) {
    int i = blockIdx.x * blockDim.x + threadIdx.x;
    if (i < qnnz) atomicAdd(&q[qi[i]], qv[i]);
}

// ---------------------------------------------------------------------------
// helper: 32-lane exact f32 sum via v_wmma_f32_16x16x4_f32 against ones.
// A(16x4): lane m (<16) -> A[m,0]=p ; lane m+16 -> A[m,2]=p ; rest 0.
// B = ones => D[m,n] = p[m] + p[m+16].  Summing the 8 D VGPRs gives the
// half-sums; one xor-16 shuffle completes the total (broadcast to all lanes).
// EXEC must be all ones at the call site.
// ---------------------------------------------------------------------------
__device__ __forceinline__ float wave_sum_wmma(float p) {
    v2f a; a.x = p;    a.y = 0.0f;
    v2f b; b.x = 1.0f; b.y = 1.0f;
    v8f c = {};
    v8f d = __builtin_amdgcn_wmma_f32_16x16x4_f32(
                false, a, false, b, (short)0, c, false, false);
    float t = ((d[0] + d[1]) + (d[2] + d[3])) +
              ((d[4] + d[5]) + (d[6] + d[7]));
    return t + __shfl_xor(t, 16, 32);
}

// ---------------------------------------------------------------------------
// 3) CSR SpMV: two docs per wave32 in flight, b128 (int4/float4) row loads
//    with scalar head/tail for 16B alignment, query gathers from LDS.
// ---------------------------------------------------------------------------
__global__ __launch_bounds__(512) void spmv_kernel(
        const long long* __restrict__ crow,
        const int* __restrict__ indice,
        const float* __restrict__ values,
        const float* __restrict__ query,
        float* __restrict__ scores,
        int nDocs) {
    __shared__ alignas(16) float sq[VOCAB];

    // vectorized LDS staging: query base is 256B aligned
    {
        const float4* q4 = (const float4*)query;
        float4* s4 = (float4*)sq;
        const int n4 = VOCAB / 4;                    // 7630
        for (int i = threadIdx.x; i < n4; i += blockDim.x) s4[i] = q4[i];
        int t = n4 * 4 + threadIdx.x;                // tail (2 elements)
        if (t < VOCAB) sq[t] = query[t];
    }
    __syncthreads();

    const int4*   indice4 = (const int4*)indice;
    const float4* values4 = (const float4*)values;

    const int lane        = threadIdx.x & 31;
    const int waveInBlock = threadIdx.x >> 5;
    const int wavesPerBlk = blockDim.x >> 5;
    const int waveId      = blockIdx.x * wavesPerBlk + waveInBlock;
    const int numWaves    = gridDim.x * wavesPerBlk;

    for (int doc = waveId; doc < nDocs; doc += 2 * numWaves) {  // wave-uniform
        const int  docA = doc;
        const int  docB = doc + numWaves;
        const bool hasB = (docB < nDocs);                       // wave-uniform

        const int sA = (int)crow[docA];
        const int eA = (int)crow[docA + 1];
        const int sB = hasB ? (int)crow[docB]     : 0;
        const int eB = hasB ? (int)crow[docB + 1] : 0;

        const int aS = (sA + 3) & ~3;   // first 16B-aligned element
        const int aE = eA & ~3;         // end of aligned region
        const int bS = (sB + 3) & ~3;
        const int bE = eB & ~3;

        float pa0 = 0.f, pa1 = 0.f, pa2 = 0.f, pa3 = 0.f;
        float pb0 = 0.f, pb1 = 0.f, pb2 = 0.f, pb3 = 0.f;

        // scalar head (<=3 elems) and tail (<=3 elems), doc A
        {
            int j = sA + lane;
            if (j < aS && j < eA) pa0 += values[j] * sq[indice[j]];
            j = aE + lane;
            if (j >= aS && j < eA) pa1 += values[j] * sq[indice[j]];
        }
        if (hasB) {
            int j = sB + lane;
            if (j < bS && j < eB) pb0 += values[j] * sq[indice[j]];
            j = bE + lane;
            if (j >= bS && j < eB) pb1 += values[j] * sq[indice[j]];
        }

        // fused vector body: both docs' b128 loads issued per iteration
        const int stA = aS >> 2, enA = aE >> 2;     // float4 chunk ranges
        const int stB = bS >> 2, enB = bE >> 2;
        int kA = stA + lane;
        int kB = stB + lane;
        const int nItA = (enA - stA + 31) >> 5;     // wave-uniform
        const int nItB = hasB ? ((enB - stB + 31) >> 5) : 0;
        const int nIt  = (nItA > nItB) ? nItA : nItB;

        for (int it = 0; it < nIt; ++it) {
            const bool aok = (kA < enA);
            const bool bok = (kB < enB);
            const int  ka  = aok ? kA : stA;        // safe clamped address
            const int  kb  = bok ? kB : stB;
            int4   ia = indice4[ka];
            float4 va = values4[ka];
            int4   ib;
            float4 vb;
            if (hasB) { ib = indice4[kb]; vb = values4[kb]; }
            if (aok) {
                pa0 += va.x * sq[ia.x];
                pa1 += va.y * sq[ia.y];
                pa2 += va.z * sq[ia.z];
                pa3 += va.w * sq[ia.w];
            }
            if (hasB && bok) {
                pb0 += vb.x * sq[ib.x];
                pb1 += vb.y * sq[ib.y];
                pb2 += vb.z * sq[ib.z];
                pb3 += vb.w * sq[ib.w];
            }
            kA += 32;
            kB += 32;
        }

        // wave reductions (EXEC all ones here; doc indices are wave-uniform)
        float totalA = wave_sum_wmma((pa0 + pa1) + (pa2 + pa3));
        if (lane == 0) scores[docA] = totalA;
        if (hasB) {
            float totalB = wave_sum_wmma((pb0 + pb1) + (pb2 + pb3));
            if (lane == 0) scores[docB] = totalB;
        }
    }
}

// ---------------------------------------------------------------------------
// top-k helpers: (value desc, index asc) ordering, matching jax.lax.top_k
// ---------------------------------------------------------------------------
__device__ __forceinline__ bool tk_better(float v1, int i1, float v2, int i2) {
    return (v1 > v2) || ((v1 == v2) && (i1 < i2));
}

// 4) per-block partial top-10 over a contiguous slice of scores
__global__ __launch_bounds__(256) void topk_partial_kernel(
        const float* __restrict__ scores, int n,
        float* __restrict__ cvals, int* __restrict__ cidx, int perBlock) {
    __shared__ float sv[256];
    __shared__ int   si[256];
    __shared__ float selV[TOPK];
    __shared__ int   selI[TOPK];

    const int begin = blockIdx.x * perBlock;
    const int end   = min(begin + perBlock, n);

    for (int k = 0; k < TOPK; ++k) {
        float bv = -INFINITY;
        int   bi = 0x7fffffff;
        for (int i = begin + (int)threadIdx.x; i < end; i += blockDim.x) {
            bool excluded = false;
            for (int t = 0; t < k; ++t) if (selI[t] == i) excluded = true;
            if (!excluded) {
                float v = scores[i];
                if (tk_better(v, i, bv, bi)) { bv = v; bi = i; }
            }
        }
        sv[threadIdx.x] = bv; si[threadIdx.x] = bi;
        __syncthreads();
        for (int off = 128; off > 0; off >>= 1) {
            if ((int)threadIdx.x < off) {
                if (tk_better(sv[threadIdx.x + off], si[threadIdx.x + off],
                              sv[threadIdx.x],       si[threadIdx.x])) {
                    sv[threadIdx.x] = sv[threadIdx.x + off];
                    si[threadIdx.x] = si[threadIdx.x + off];
                }
            }
            __syncthreads();
        }
        if (threadIdx.x == 0) { selV[k] = sv[0]; selI[k] = si[0]; }
        __syncthreads();
    }
    if (threadIdx.x < TOPK) {
        cvals[blockIdx.x * TOPK + threadIdx.x] = selV[threadIdx.x];
        cidx [blockIdx.x * TOPK + threadIdx.x] = selI[threadIdx.x];
    }
}

// 5) final top-10 over NB_TOPK*10 candidates; tie-break by doc index,
//    exclusion by candidate position. Writes [10 values | 10 indices-as-float].
__global__ __launch_bounds__(256) void topk_final_kernel(
        const float* __restrict__ cvals, const int* __restrict__ cidx,
        int nCand, float* __restrict__ out) {
    __shared__ float sv[256];
    __shared__ int   sp[256];
    __shared__ int   selPos[TOPK];
    __shared__ float selV[TOPK];
    __shared__ int   selDoc[TOPK];

    for (int k = 0; k < TOPK; ++k) {
        float bv = -INFINITY;
        int   bdoc = 0x7fffffff;
        int   bpos = -1;
        for (int i = (int)threadIdx.x; i < nCand; i += blockDim.x) {
            bool excluded = false;
            for (int t = 0; t < k; ++t) if (selPos[t] == i) excluded = true;
            if (!excluded) {
                float v = cvals[i];
                int   dci = cidx[i];
                if (tk_better(v, dci, bv, bdoc)) { bv = v; bdoc = dci; bpos = i; }
            }
        }
        sv[threadIdx.x] = bv; sp[threadIdx.x] = bpos;
        __syncthreads();
        for (int off = 128; off > 0; off >>= 1) {
            if ((int)threadIdx.x < off) {
                float ov = sv[threadIdx.x + off];
                int   op = sp[threadIdx.x + off];
                int   od = (op >= 0) ? cidx[op] : 0x7fffffff;
                int   mp = sp[threadIdx.x];
                int   md = (mp >= 0) ? cidx[mp] : 0x7fffffff;
                if (tk_better(ov, od, sv[threadIdx.x], md)) {
                    sv[threadIdx.x] = ov; sp[threadIdx.x] = op;
                }
            }
            __syncthreads();
        }
        if (threadIdx.x == 0) {
            int p = sp[0];
            selPos[k] = p;
            selV[k]   = sv[0];
            selDoc[k] = (p >= 0) ? cidx[p] : 0;
        }
        __syncthreads();
    }
    if (threadIdx.x < TOPK) {
        out[threadIdx.x]        = selV[threadIdx.x];
        out[TOPK + threadIdx.x] = (float)selDoc[threadIdx.x];
    }
}

// ---------------------------------------------------------------------------
// launcher
// ---------------------------------------------------------------------------
extern "C" void kernel_launch(void* const* d_in, const int* in_sizes, int n_in,
                              void* d_out, int out_size, void* d_ws, size_t ws_size,
                              hipStream_t stream) {
    const int*       q_indices = (const int*)d_in[0];
    const float*     q_values  = (const float*)d_in[1];
    const long long* crow      = (const long long*)d_in[2];
    const int*       indice    = (const int*)d_in[3];
    const float*     values    = (const float*)d_in[4];
    // d_in[5] = top_k scalar (known constant 10)

    const int qnnz  = in_sizes[0];
    const int nDocs = in_sizes[2] - 1;

    // workspace layout
    char* ws = (char*)d_ws;
    float* query  = (float*)ws;                                   // VOCAB floats
    size_t off = ((size_t)VOCAB * sizeof(float) + 255) & ~(size_t)255;
    float* scores = (float*)(ws + off);                           // nDocs floats
    off += (((size_t)nDocs * sizeof(float)) + 255) & ~(size_t)255;
    float* cvals  = (float*)(ws + off);                           // NB_TOPK*TOPK
    off += ((size_t)NB_TOPK * TOPK * sizeof(float) + 255) & ~(size_t)255;
    int*   cidx   = (int*)(ws + off);                             // NB_TOPK*TOPK

    // 1) zero dense query
    zero_query_kernel<<<(VOCAB + 255) / 256, 256, 0, stream>>>(query);
    // 2) scatter query nnz
    scatter_query_kernel<<<(qnnz + 63) / 64, 64, 0, stream>>>(q_indices, q_values,
                                                              query, qnnz);
    // 3) SpMV: 512-thread blocks (16 waves), two docs in flight per wave
    spmv_kernel<<<1024, 512, 0, stream>>>(crow, indice, values, query, scores, nDocs);
    // 4) partial top-k
    int perBlock = (nDocs + NB_TOPK - 1) / NB_TOPK;
    topk_partial_kernel<<<NB_TOPK, 256, 0, stream>>>(scores, nDocs, cvals, cidx,
                                                     perBlock);
    // 5) final top-k -> d_out
    topk_final_kernel<<<1, 256, 0, stream>>>(cvals, cidx, NB_TOPK * TOPK,
                                             (float*)d_out);
}